// LDPCBeliefPropagation_72258529788541
// MI455X (gfx1250) — compile-verified
//
#include <hip/hip_runtime.h>

// ---------------------------------------------------------------------------
// LDPC sum-product BP, (7,4) Hamming graph, B codewords x 5 iterations.
// TRANS-bound => gfx1250 v_tanh_f32 / v_rcp_f32 / v_log_f32, 12 TRANS per
// check row via (t_e - T)/(t_e + T) identity (exact rewrite of the
// log-sum form; removes the first log layer entirely).
// 28B-stride per-codeword layout => async global->LDS staging (CDNA5 path),
// per-wave private regions, no workgroup barrier, conflict-free LDS reads.
// ---------------------------------------------------------------------------

#ifndef __has_builtin
#define __has_builtin(x) 0
#endif

#define GAS __attribute__((address_space(1)))
#define LAS __attribute__((address_space(3)))

#if __has_builtin(__builtin_amdgcn_global_load_async_to_lds_b32)
#define HAVE_ASYNC_LDS_BUILTIN 1
#else
#define HAVE_ASYNC_LDS_BUILTIN 0
#endif

#if __has_builtin(__builtin_amdgcn_s_wait_asynccnt)
#define WAIT_ASYNC0() do { __builtin_amdgcn_s_wait_asynccnt(0); \
                           asm volatile("" ::: "memory"); } while (0)
#else
#define WAIT_ASYNC0() asm volatile("s_wait_asynccnt 0x0" ::: "memory")
#endif

#if __has_builtin(__builtin_amdgcn_tanhf)
#define FAST_TANH(x) __builtin_amdgcn_tanhf(x)   // v_tanh_f32 (CDNA5 TRANS)
#else
#define FAST_TANH(x) tanhf(x)
#endif

#if __has_builtin(__builtin_amdgcn_logf)
#define FAST_LOG2(x) __builtin_amdgcn_logf(x)    // v_log_f32
#else
#define FAST_LOG2(x) __log2f(x)
#endif

#if __has_builtin(__builtin_amdgcn_rcpf)
#define FAST_RCP(x) __builtin_amdgcn_rcpf(x)     // v_rcp_f32
#else
#define FAST_RCP(x) (1.0f / (x))
#endif

// jnp.sign semantics: sign(0) = 0
__device__ __forceinline__ float sgn_f(float x) {
  return (x > 0.0f) ? 1.0f : ((x < 0.0f) ? -1.0f : 0.0f);
}

// Check-node update over 4 incident edges.
// Reference: lu_e = log|tanh(m_e/2)|; s_e = sum(lu) - lu_e;
//            c2v_e = -log|tanh(s_e/2)| * prod_{e'!=e} sign(m_{e'})
// Identity:  with t_e = |tanh(m_e/2)|, T = prod(t),
//            |tanh(s_e/2)| = (t_e - T)/(t_e + T)   (since s_e = log(T/t_e))
// => 4 tanh + 4 rcp + 4 log = 12 TRANS/row (vs 16 for the log-sum form).
__device__ __forceinline__ void check_row(float a, float b, float c, float d,
                                          float* o) {
  const float LN2 = 0.69314718055994530942f;
  float ta = fabsf(FAST_TANH(0.5f * a));
  float tb = fabsf(FAST_TANH(0.5f * b));
  float tc = fabsf(FAST_TANH(0.5f * c));
  float td = fabsf(FAST_TANH(0.5f * d));
  float T = (ta * tb) * (tc * td);
  float ra = (ta - T) * FAST_RCP(ta + T);
  float rb = (tb - T) * FAST_RCP(tb + T);
  float rc = (tc - T) * FAST_RCP(tc + T);
  float rd = (td - T) * FAST_RCP(td + T);
  float pa = LN2 * FAST_LOG2(ra);   // = phi(s_a) <= 0
  float pb = LN2 * FAST_LOG2(rb);
  float pc = LN2 * FAST_LOG2(rc);
  float pd = LN2 * FAST_LOG2(rd);
  float sa = sgn_f(a), sb = sgn_f(b), sc = sgn_f(c), sd = sgn_f(d);
  float pr = (sa * sb) * (sc * sd);  // row sign product
  // pr / s_e == pr * s_e for s_e in {+1,-1}
  o[0] = -pa * (pr * sa);
  o[1] = -pb * (pr * sb);
  o[2] = -pc * (pr * sc);
  o[3] = -pd * (pr * sd);
}

__global__ __launch_bounds__(256) void ldpc_bp_kernel(
    const float* __restrict__ llr, const int* __restrict__ p_iters,
    float* __restrict__ out, int B) {
  __shared__ float stage[256 * 7];  // 7168 B: per-block input staging

  const int t    = threadIdx.x;
  const int lane = t & 31;
  const int wave = t >> 5;
  const int cw0  = blockIdx.x * 256;
  const int cw   = cw0 + t;
  const int niters = p_iters[0];  // uniform scalar load

  float L[7];
  const bool fullblock = (cw0 + 256) <= B;
  if (fullblock) {
    // Wave-private region: 32 codewords * 7 floats = 224 words = 896 bytes.
    // 7 coalesced async b32 loads per lane; INST_OFFSET advances global and
    // LDS addresses together, so the k-th load moves word (lane + 32k).
    const float* gw = llr + (size_t)(cw0 + wave * 32) * 7;
    float* sw = &stage[wave * 32 * 7];
#if HAVE_ASYNC_LDS_BUILTIN
    GAS int* g = (GAS int*)(gw + lane);
    LAS int* l = (LAS int*)(sw + lane);
    __builtin_amdgcn_global_load_async_to_lds_b32(g, l, 0, 0);
    __builtin_amdgcn_global_load_async_to_lds_b32(g, l, 128, 0);
    __builtin_amdgcn_global_load_async_to_lds_b32(g, l, 256, 0);
    __builtin_amdgcn_global_load_async_to_lds_b32(g, l, 384, 0);
    __builtin_amdgcn_global_load_async_to_lds_b32(g, l, 512, 0);
    __builtin_amdgcn_global_load_async_to_lds_b32(g, l, 640, 0);
    __builtin_amdgcn_global_load_async_to_lds_b32(g, l, 768, 0);
#else
    unsigned long long ga = (unsigned long long)(uintptr_t)(gw + lane);
    unsigned ldso = (unsigned)(uintptr_t)(LAS int*)(sw + lane);
    asm volatile(
        "global_load_async_to_lds_b32 %0, %1, off\n\t"
        "global_load_async_to_lds_b32 %0, %1, off offset:128\n\t"
        "global_load_async_to_lds_b32 %0, %1, off offset:256\n\t"
        "global_load_async_to_lds_b32 %0, %1, off offset:384\n\t"
        "global_load_async_to_lds_b32 %0, %1, off offset:512\n\t"
        "global_load_async_to_lds_b32 %0, %1, off offset:640\n\t"
        "global_load_async_to_lds_b32 %0, %1, off offset:768"
        :: "v"(ldso), "v"(ga) : "memory");
#endif
    WAIT_ASYNC0();  // only this wave's ASYNCcnt; region is wave-private
    // Conflict-free: lane reads words 7*lane..7*lane+6; 7 coprime to 64 banks.
    const float* mine = &sw[lane * 7];
#pragma unroll
    for (int j = 0; j < 7; ++j) L[j] = mine[j];
  } else {
    if (cw >= B) return;
#pragma unroll
    for (int j = 0; j < 7; ++j) L[j] = llr[(size_t)cw * 7 + j];
  }

  // Variable->check messages on the 12 edges, init = H * llr.
  float m0 = L[0], m1 = L[2], m2 = L[4], m3 = L[6];   // check 0: v 0,2,4,6
  float m4 = L[1], m5 = L[2], m6 = L[5], m7 = L[6];   // check 1: v 1,2,5,6
  float m8 = L[3], m9 = L[4], m10 = L[5], m11 = L[6]; // check 2: v 3,4,5,6

  float NL[7] = {L[0], L[1], L[2], L[3], L[4], L[5], L[6]};

  // Global early-exit flag in the reference is all(synd==0) over the ENTIRE
  // batch -> false w.p. ~1 for Gaussian LLRs and never affects llr_tot, so we
  // run all iterations unconditionally (per-codeword exit would be unfaithful).
  for (int it = 0; it < niters; ++it) {
    float c0[4], c1[4], c2[4];
    check_row(m0, m1, m2, m3, c0);
    check_row(m4, m5, m6, m7, c1);
    check_row(m8, m9, m10, m11, c2);

    NL[0] = L[0] + c0[0];
    NL[1] = L[1] + c1[0];
    NL[2] = L[2] + c0[1] + c1[1];
    NL[3] = L[3] + c2[0];
    NL[4] = L[4] + c0[2] + c2[1];
    NL[5] = L[5] + c1[2] + c2[2];
    NL[6] = L[6] + (c0[3] + c1[3]) + c2[3];

    // msgs = H*new_llr - c2v (extrinsic)
    m0 = NL[0] - c0[0]; m1 = NL[2] - c0[1]; m2 = NL[4] - c0[2]; m3 = NL[6] - c0[3];
    m4 = NL[1] - c1[0]; m5 = NL[2] - c1[1]; m6 = NL[5] - c1[2]; m7 = NL[6] - c1[3];
    m8 = NL[3] - c2[0]; m9 = NL[4] - c2[1]; m10 = NL[5] - c2[2]; m11 = NL[6] - c2[3];
  }

  // Direct stores: compiler merges the 7 consecutive floats per lane into
  // b128+b96; kernel is TRANS-bound so store pattern is off the critical path.
  float* o = out + (size_t)cw * 7;
#pragma unroll
  for (int j = 0; j < 7; ++j) o[j] = NL[j];
}

extern "C" void kernel_launch(void* const* d_in, const int* in_sizes, int n_in,
                              void* d_out, int out_size, void* d_ws,
                              size_t ws_size, hipStream_t stream) {
  const float* llr = (const float*)d_in[0];
  const int* max_iters = (const int*)d_in[1];  // device scalar (==5)
  int B = in_sizes[0] / 7;                     // 262144 codewords
  int blocks = (B + 255) / 256;
  ldpc_bp_kernel<<<blocks, 256, 0, stream>>>(llr, max_iters, (float*)d_out, B);
}